// RoIAlign_79207786873466
// MI455X (gfx1250) — compile-verified
//
#include <hip/hip_runtime.h>

// RoIAlign (K=1000 rois, feat 2x256x200x200 f32, out 1000x256x7x7 f32)
// Strategy for MI455X:
//  1) feat (80 MB) fits in the 192 MB L2; op is gather-bound (~250 MFLOP total).
//  2) Pass 1: NCHW -> NHWC transpose into d_ws so channel gathers are contiguous
//     (1 KB per (y,x)) -> coalesced wave loads instead of 32-line scatter.
//  3) Pass 2: per-ROI blocks. Each wave computes a 16-channel x 14-sample
//     y-combined/x-interpolated matrix Q and contracts it with the constant
//     0.5-average pooling matrix via a chain of V_WMMA_F32_16X16X4_F32
//     (M=16 ch, N=16>=7 bins, K=4), exact f32 numerics.

typedef __attribute__((ext_vector_type(2))) float v2f;
typedef __attribute__((ext_vector_type(8))) float v8f;

#define RH 200
#define RW 200
#define RC 256
#define OH 7
#define OW 7
#define NS 14   /* OH*SR = OW*SR sample count per axis */

// ---------------- Pass 1: NCHW -> NHWC tiled transpose ----------------
__global__ __launch_bounds__(256) void nchw_to_nhwc(const float* __restrict__ in,
                                                    float* __restrict__ out) {
    __shared__ float tile[32][33];
    const int p0 = blockIdx.x * 32;           // spatial (y*W+x) tile
    const int c0 = blockIdx.y * 32;           // channel tile
    const int n  = blockIdx.z;
    const int tx = threadIdx.x, ty = threadIdx.y;
    const size_t HW = (size_t)RH * RW;
#pragma unroll
    for (int j = 0; j < 32; j += 8)
        tile[ty + j][tx] = in[((size_t)(n * RC + c0 + ty + j)) * HW + (p0 + tx)];
    __syncthreads();
#pragma unroll
    for (int j = 0; j < 32; j += 8)
        out[((size_t)n * HW + (p0 + ty + j)) * RC + (c0 + tx)] = tile[tx][ty + j];
}

// ---------------- Pass 2: RoIAlign with WMMA pooling ----------------
__global__ __launch_bounds__(256) void roi_align_wmma(const float* __restrict__ featT,
                                                      const float* __restrict__ rois,
                                                      float* __restrict__ out) {
    __shared__ int   s_ylo[NS], s_yhi[NS];
    __shared__ float s_wy0[NS], s_wy1[NS];      // include validity + 0.5 (y-avg)
    __shared__ int   s_xlo[16], s_xhi[16];      // padded to 16 for K=16 chain
    __shared__ float s_wx0[16], s_wx1[16];      // include validity
    __shared__ int   s_n;

    const int k   = blockIdx.x;
    const int tid = threadIdx.x;

    // --- per-ROI geometry (mirrors _sample_coords/_bilinear_prep) ---
    if (tid < 2 * NS) {
        const bool isY = tid < NS;
        const int  i   = isY ? tid : tid - NS;
        const float start = rois[5 * k + (isY ? 1 : 2)] * 0.25f;
        const float end   = rois[5 * k + (isY ? 3 : 4)] * 0.25f;
        const float bin   = fmaxf(end - start, 1.0f) * (1.0f / 7.0f);
        const float frac  = (float)(i >> 1) + ((float)(i & 1) + 0.5f) * 0.5f;
        const float coord = start + bin * frac;
        const bool  valid = (coord >= -1.0f) && (coord <= 200.0f);
        float cc = fmaxf(coord, 0.0f);
        int   lo0 = (int)floorf(cc);
        const bool at_edge = lo0 >= 199;
        int lo = at_edge ? 199 : lo0;
        int hi = at_edge ? 199 : lo0 + 1;
        cc = at_edge ? 199.0f : cc;
        lo = min(max(lo, 0), 199);
        hi = min(max(hi, 0), 199);
        const float l = cc - (float)lo;
        const float h = 1.0f - l;
        const float wlo = valid ? h : 0.0f;
        const float whi = valid ? l : 0.0f;
        if (isY) { s_ylo[i] = lo; s_yhi[i] = hi; s_wy0[i] = 0.5f * wlo; s_wy1[i] = 0.5f * whi; }
        else     { s_xlo[i] = lo; s_xhi[i] = hi; s_wx0[i] = wlo;        s_wx1[i] = whi; }
    } else if (tid == 2 * NS) {
        s_n = (int)rois[5 * k];
        s_xlo[14] = 0; s_xhi[14] = 0; s_wx0[14] = 0.0f; s_wx1[14] = 0.0f;
        s_xlo[15] = 0; s_xhi[15] = 0; s_wx0[15] = 0.0f; s_wx1[15] = 0.0f;
    }
    __syncthreads();

    const int wave = tid >> 5;
    const int lane = tid & 31;
    const int m    = lane & 15;   // M row (channel-in-tile) / N col (pw)
    const int half = lane >> 4;   // selects K columns 0,1 vs 2,3 of A/B
    const int n_img = s_n;

    // 7 bins-in-y x 16 channel tiles = 112 wave tasks over 8 waves
    for (int task = wave; task < OH * (RC / 16); task += 8) {
        const int ph = task >> 4;
        const int ct = task & 15;
        const int c  = ct * 16 + m;

        const int j0 = 2 * ph, j1 = j0 + 1;
        const int   ry0 = s_ylo[j0], ry1 = s_yhi[j0], ry2 = s_ylo[j1], ry3 = s_yhi[j1];
        const float wy0 = s_wy0[j0], wy1 = s_wy1[j0], wy2 = s_wy0[j1], wy3 = s_wy1[j1];
        const size_t nb = (size_t)n_img * (RH * RW);
        const size_t rb0 = (nb + (size_t)ry0 * RW) * RC + c;
        const size_t rb1 = (nb + (size_t)ry1 * RW) * RC + c;
        const size_t rb2 = (nb + (size_t)ry2 * RW) * RC + c;
        const size_t rb3 = (nb + (size_t)ry3 * RW) * RC + c;

        v8f acc = {};
#pragma unroll
        for (int s = 0; s < 4; ++s) {
            // ISA 16x4 f32 A layout: VGPR0 = K {0 | 2}, VGPR1 = K {1 | 3} per lane half
            const int k0 = 4 * s + 2 * half;
            const int k1 = k0 + 1;

            const int   oa0 = s_xlo[k0] * RC, oa1 = s_xhi[k0] * RC;
            const float wa0 = s_wx0[k0],      wa1 = s_wx1[k0];
            const float a0 =
                  wy0 * (wa0 * featT[rb0 + oa0] + wa1 * featT[rb0 + oa1])
                + wy1 * (wa0 * featT[rb1 + oa0] + wa1 * featT[rb1 + oa1])
                + wy2 * (wa0 * featT[rb2 + oa0] + wa1 * featT[rb2 + oa1])
                + wy3 * (wa0 * featT[rb3 + oa0] + wa1 * featT[rb3 + oa1]);

            const int   ob0 = s_xlo[k1] * RC, ob1 = s_xhi[k1] * RC;
            const float wb0 = s_wx0[k1],      wb1 = s_wx1[k1];
            const float a1 =
                  wy0 * (wb0 * featT[rb0 + ob0] + wb1 * featT[rb0 + ob1])
                + wy1 * (wb0 * featT[rb1 + ob0] + wb1 * featT[rb1 + ob1])
                + wy2 * (wb0 * featT[rb2 + ob0] + wb1 * featT[rb2 + ob1])
                + wy3 * (wb0 * featT[rb3 + ob0] + wb1 * featT[rb3 + ob1]);

            // Constant pooling matrix B[sx, pw] = 0.5 iff pw == sx/2, sx < 14
            const float b0 = ((k0 < NS) && (m == (k0 >> 1))) ? 0.5f : 0.0f;
            const float b1 = ((k1 < NS) && (m == (k1 >> 1))) ? 0.5f : 0.0f;

            v2f A = {a0, a1};
            v2f B = {b0, b1};
            acc = __builtin_amdgcn_wmma_f32_16x16x4_f32(
                /*neg_a=*/false, A, /*neg_b=*/false, B,
                /*c_mod=*/(short)0, acc, /*reuse_a=*/false, /*reuse_b=*/false);
        }

        // D layout: VGPR v holds row M = v + half*8, col N = m (= pw)
        if (m < OW) {
            const int pw = m;
#pragma unroll
            for (int v = 0; v < 8; ++v) {
                const int cc = ct * 16 + half * 8 + v;
                out[(((size_t)k * RC + cc) * OH + ph) * OW + pw] = acc[v];
            }
        }
    }
}

extern "C" void kernel_launch(void* const* d_in, const int* in_sizes, int n_in,
                              void* d_out, int out_size, void* d_ws, size_t ws_size,
                              hipStream_t stream) {
    const float* feat = (const float*)d_in[0];   // (2,256,200,200) f32
    const float* rois = (const float*)d_in[1];   // (K,5) f32
    float* out  = (float*)d_out;                 // (K,256,7,7) f32
    float* featT = (float*)d_ws;                 // NHWC scratch: 2*200*200*256*4 = 81,920,000 B

    const int K = in_sizes[1] / 5;

    dim3 tb(32, 8);
    dim3 tg((RH * RW) / 32, RC / 32, 2);         // 1250 x 8 x 2
    nchw_to_nhwc<<<tg, tb, 0, stream>>>(feat, featT);

    roi_align_wmma<<<dim3(K), 256, 0, stream>>>(featT, rois, out);
    (void)out_size; (void)ws_size; (void)n_in;
}